// MultiSimilarityLoss_395136991969
// MI455X (gfx1250) — compile-verified
//
#include <hip/hip_runtime.h>
#include <math.h>

// ---------------------------------------------------------------------------
// Multi-Similarity loss for MI455X (gfx1250, wave32).
//   feats = concat(e1,e2) : 4096 x 512 fp32 (rows L2-normalized)
//   sim   = feats @ feats^T computed on the fp32 matrix pipe
//           (v_wmma_f32_16x16x4_f32), never materialized in memory.
// Two passes over the Gram matrix (mining needs full rows first):
//   pass 1: per-row masked min(pos) / max(neg)   -> 32 KB workspace
//   pass 2: recompute tiles, mine pairs, log1p(sum exp), atomic scalar sum
// ---------------------------------------------------------------------------

typedef float v2f __attribute__((ext_vector_type(2)));
typedef float v8f __attribute__((ext_vector_type(8)));

#define N_TOT 4096
#define HALFN 2048
#define DDIM  512
#define LDSPITCH (DDIM + 4)   // pad: 4*m mod 64 distinct banks for m=0..15

#define F_SCALE_POS 2.0f
#define F_SCALE_NEG 40.0f
#define F_THRESH    0.5f
#define F_MARGIN    0.1f
#define F_EPS       1e-5f

__device__ __forceinline__ const float* feat_row(const float* __restrict__ e1,
                                                 const float* __restrict__ e2,
                                                 int i) {
    return (i < HALFN) ? (e1 + (size_t)i * DDIM)
                       : (e2 + (size_t)(i - HALFN) * DDIM);
}
__device__ __forceinline__ int lbl_at(const int* __restrict__ l1,
                                      const int* __restrict__ l2, int i) {
    return (i < HALFN) ? l1[i] : l2[i - HALFN];
}

// Compute one 16x16 fp32 Gram tile: C = A(16xK) * B(Kx16), K = 512.
//   lane L: m/n = L&15, kh = L>>4
//   A vregs: {F[row+m][k+2kh], F[row+m][k+2kh+1]}   (from LDS panel)
//   B vregs: {F[col+n][k+2kh], F[col+n][k+2kh+1]}   (from global/L2)
__device__ __forceinline__ v8f gram_tile_f32(const float* __restrict__ aPtr,
                                             const float* __restrict__ bPtr) {
    v8f c = {0.f, 0.f, 0.f, 0.f, 0.f, 0.f, 0.f, 0.f};
#pragma unroll 8
    for (int k = 0; k < DDIM; k += 4) {
        v2f a = *(const v2f*)(aPtr + k);
        v2f b = *(const v2f*)(bPtr + k);
        c = __builtin_amdgcn_wmma_f32_16x16x4_f32(false, a, false, b,
                                                  (short)0, c, false, false);
    }
    return c;
}

// ---------------------------------------------------------------------------
// Pass 1: per-row hardest-positive (min over positives) / hardest-negative
//         (max over negatives) thresholds.
// ---------------------------------------------------------------------------
__global__ __launch_bounds__(256) void ms_pass_minmax(
        const float* __restrict__ e1, const float* __restrict__ e2,
        const int* __restrict__ l1, const int* __restrict__ l2,
        float* __restrict__ minPos, float* __restrict__ maxNeg,
        float* __restrict__ out)
{
    __shared__ float ldsA[16 * LDSPITCH];
    __shared__ int   rowLab[16];
    __shared__ float redMin[8][16];
    __shared__ float redMax[8][16];

    const int tid  = threadIdx.x;
    const int wave = tid >> 5;
    const int lane = tid & 31;
    const int mn_  = lane & 15;   // A row within tile / B col within tile
    const int kh   = lane >> 4;   // k-half for A/B, row-half for C
    const int rowBase = blockIdx.x * 16;

    if (blockIdx.x == 0 && tid == 0) out[0] = 0.f;   // harness poisons d_out

    for (int idx = tid; idx < 16 * DDIM; idx += 256) {
        const int r = idx >> 9;            // / 512
        const int c = idx & (DDIM - 1);
        ldsA[r * LDSPITCH + c] = feat_row(e1, e2, rowBase + r)[c];
    }
    if (tid < 16) rowLab[tid] = lbl_at(l1, l2, rowBase + tid);
    __syncthreads();

    float accMin[8], accMax[8];
#pragma unroll
    for (int r = 0; r < 8; ++r) { accMin[r] = INFINITY; accMax[r] = -INFINITY; }

    const float* aPtr = &ldsA[mn_ * LDSPITCH + 2 * kh];

    for (int ct = wave; ct < N_TOT / 16; ct += 8) {
        const int colBase = ct * 16;
        const float* bPtr = feat_row(e1, e2, colBase + mn_) + 2 * kh;
        const int   lc    = lbl_at(l1, l2, colBase + mn_);

        v8f c = gram_tile_f32(aPtr, bPtr);

#pragma unroll
        for (int r = 0; r < 8; ++r) {
            const float s  = c[r];
            const int   lr = rowLab[r + 8 * kh];
            if (lr == lc) {
                if (s < 1.f - F_EPS) accMin[r] = fminf(accMin[r], s);
            } else {
                accMax[r] = fmaxf(accMax[r], s);
            }
        }
    }

    // reduce the 16 columns of the tile across each 16-lane half (wave32)
#pragma unroll
    for (int r = 0; r < 8; ++r) {
        float v0 = accMin[r], v1 = accMax[r];
        for (int o = 8; o; o >>= 1) {
            v0 = fminf(v0, __shfl_xor(v0, o, 16));
            v1 = fmaxf(v1, __shfl_xor(v1, o, 16));
        }
        if (mn_ == 0) {
            redMin[wave][r + 8 * kh] = v0;
            redMax[wave][r + 8 * kh] = v1;
        }
    }
    __syncthreads();

    if (tid < 16) {
        float mn = INFINITY, mx = -INFINITY;
#pragma unroll
        for (int w = 0; w < 8; ++w) {
            mn = fminf(mn, redMin[w][tid]);
            mx = fmaxf(mx, redMax[w][tid]);
        }
        minPos[rowBase + tid] = mn;   // +inf when no positive -> invalid row
        maxNeg[rowBase + tid] = mx;   // -inf when no negative -> invalid row
    }
}

// ---------------------------------------------------------------------------
// Pass 2: recompute tiles, mine pairs against the thresholds, accumulate
//         masked sum-exp per row, emit scalar loss.
// ---------------------------------------------------------------------------
__global__ __launch_bounds__(256) void ms_pass_loss(
        const float* __restrict__ e1, const float* __restrict__ e2,
        const int* __restrict__ l1, const int* __restrict__ l2,
        const float* __restrict__ minPos, const float* __restrict__ maxNeg,
        float* __restrict__ out)
{
    __shared__ float ldsA[16 * LDSPITCH];
    __shared__ int   rowLab[16];
    __shared__ float rowMinPos[16];
    __shared__ float rowMaxNeg[16];
    __shared__ float redPos[8][16];
    __shared__ float redNeg[8][16];

    const int tid  = threadIdx.x;
    const int wave = tid >> 5;
    const int lane = tid & 31;
    const int mn_  = lane & 15;
    const int kh   = lane >> 4;
    const int rowBase = blockIdx.x * 16;

    for (int idx = tid; idx < 16 * DDIM; idx += 256) {
        const int r = idx >> 9;
        const int c = idx & (DDIM - 1);
        ldsA[r * LDSPITCH + c] = feat_row(e1, e2, rowBase + r)[c];
    }
    if (tid < 16) {
        rowLab[tid]    = lbl_at(l1, l2, rowBase + tid);
        rowMinPos[tid] = minPos[rowBase + tid];
        rowMaxNeg[tid] = maxNeg[rowBase + tid];
    }
    __syncthreads();

    float accP[8], accN[8];
#pragma unroll
    for (int r = 0; r < 8; ++r) { accP[r] = 0.f; accN[r] = 0.f; }

    const float* aPtr = &ldsA[mn_ * LDSPITCH + 2 * kh];

    for (int ct = wave; ct < N_TOT / 16; ct += 8) {
        const int colBase = ct * 16;
        const float* bPtr = feat_row(e1, e2, colBase + mn_) + 2 * kh;
        const int   lc    = lbl_at(l1, l2, colBase + mn_);

        v8f c = gram_tile_f32(aPtr, bPtr);

#pragma unroll
        for (int r = 0; r < 8; ++r) {
            const float s   = c[r];
            const int   row = r + 8 * kh;
            const int   lr  = rowLab[row];
            if (lr == lc) {
                // pos_sel = pos_mask & (sim - MARGIN < max_neg)
                if (s < 1.f - F_EPS && (s - F_MARGIN) < rowMaxNeg[row])
                    accP[r] += expf(-F_SCALE_POS * (s - F_THRESH));
            } else {
                // neg_sel = neg_mask & (sim + MARGIN > min_pos)
                if ((s + F_MARGIN) > rowMinPos[row])
                    accN[r] += expf(F_SCALE_NEG * (s - F_THRESH));
            }
        }
    }

#pragma unroll
    for (int r = 0; r < 8; ++r) {
        float p = accP[r], n = accN[r];
        for (int o = 8; o; o >>= 1) {
            p += __shfl_xor(p, o, 16);
            n += __shfl_xor(n, o, 16);
        }
        if (mn_ == 0) {
            redPos[wave][r + 8 * kh] = p;
            redNeg[wave][r + 8 * kh] = n;
        }
    }
    __syncthreads();

    if (tid < 16) {
        float p = 0.f, n = 0.f;
#pragma unroll
        for (int w = 0; w < 8; ++w) { p += redPos[w][tid]; n += redNeg[w][tid]; }
        // selected exponentials are strictly > 0, so sum>0 <=> any selected
        const bool valid = (p > 0.f) && (n > 0.f);
        const float loss = valid
            ? (log1pf(p) * (1.f / F_SCALE_POS) + log1pf(n) * (1.f / F_SCALE_NEG))
            : 0.f;
        atomicAdd(out, loss * (1.f / (float)N_TOT));
    }
}

// ---------------------------------------------------------------------------
extern "C" void kernel_launch(void* const* d_in, const int* in_sizes, int n_in,
                              void* d_out, int out_size, void* d_ws, size_t ws_size,
                              hipStream_t stream) {
    const float* e1 = (const float*)d_in[0];   // embedding1 (2048x512 f32)
    const float* e2 = (const float*)d_in[1];   // embedding2 (2048x512 f32)
    const int*   l1 = (const int*)d_in[2];     // label1 (2048 i32)
    const int*   l2 = (const int*)d_in[3];     // label2 (2048 i32)
    float* out = (float*)d_out;                // scalar f32 loss

    float* minPos = (float*)d_ws;              // 4096 f32
    float* maxNeg = minPos + N_TOT;            // 4096 f32 (32 KB total)

    ms_pass_minmax<<<N_TOT / 16, 256, 0, stream>>>(e1, e2, l1, l2,
                                                   minPos, maxNeg, out);
    ms_pass_loss  <<<N_TOT / 16, 256, 0, stream>>>(e1, e2, l1, l2,
                                                   minPos, maxNeg, out);
}